// SpectralFractionalConv3d_84267258347792
// MI455X (gfx1250) — compile-verified
//
#include <hip/hip_runtime.h>
#include <math.h>

// ============================================================================
// SpectralFractionalConv3d on MI455X (gfx1250, wave32)
//
// Pipeline (all complex fp32, v_wmma_f32_16x16x4_f32), planar re/im storage
// for all intermediates (zero deinterleave movs feeding WMMA):
//   x --F(a1,W)--> dout --F(a2,H)--> ws --F(a3,D)--> dout --Wc+b(C)--> ws
//     --F(-a3,D)--> dout --F(-a2,H)--> ws --F(-a1,W)--> dout (interleaved c64)
//
// ~56 GFLOP total, ~1.9 GB traffic -> bandwidth bound at 23.3 TB/s (mostly
// L2-resident: two 134 MB ping-pong buffers vs 192 MB L2). fp32 WMMA keeps
// the 6 chained spectral transforms numerically tight; lower precision buys
// nothing on a memory-bound op.
//
// F matrices are stored planar [re(KD*KD) | im(KD*KD)] so the per-block
// F -> LDS staging is one contiguous copy, issued via the Tensor Data Mover
// (tensor_load_to_lds + s_wait_tensorcnt; 6-arg clang-23 builtin form).
//
// Workspace (~135.3 MB):
//   [0,196608)        : 6 transform matrices, planar complex 64x64
//   [196608,204800)   : Wc 32x32 planar complex
//   [204800,205056)   : bias 32 planar complex
//   [205056,221440)   : E basis 64x64 fp32
//   [1MB, 1MB+134MB)  : planar ping buffer (re | im planes)
// ============================================================================

typedef __attribute__((ext_vector_type(2))) float v2f;
typedef __attribute__((ext_vector_type(8))) float v8f;
typedef __attribute__((ext_vector_type(4))) unsigned int v4u;
typedef __attribute__((ext_vector_type(8))) int v8i;
typedef __attribute__((ext_vector_type(4))) int v4i;

// ---------------------------------------------------------------------------
// Setup kernel 1: build the DFT-commuting eigenbasis E (64x64), alpha-free.
// Block 0: even-symmetry block (33x33), block 1: odd block (31x31).
// Cyclic Jacobi; eigenpairs sorted descending. Column signs are irrelevant
// (F = E diag Ph E^T is sign-invariant per column).
// ---------------------------------------------------------------------------
__global__ __launch_bounds__(64) void frft_build_basis(float* __restrict__ gE) {
  __shared__ float smS[64 * 64];   // S, then CS = P S P^T
  __shared__ float smP[64 * 64];
  __shared__ float smT[64 * 64];   // T = P S
  __shared__ float A[33 * 33];
  __shared__ float V[33 * 33];
  __shared__ float rot_c, rot_s;
  __shared__ int ord[33];

  const int tid = threadIdx.x;
  const int NJ = (blockIdx.x == 0) ? 33 : 31;
  const int r0 = (blockIdx.x == 0) ? 0 : 33;

  for (int t = tid; t < 4096; t += 64) {
    int i = t >> 6, j = t & 63;
    int d = (i - j) & 63;
    float h = (d == 0) ? -2.0f : ((d == 1 || d == 63) ? 1.0f : 0.0f);
    float dg = (i == j) ? (2.0f * cosf(6.283185307179586f * (float)i / 64.0f) - 2.0f) : 0.0f;
    smS[t] = h + dg;
  }
  for (int t = tid; t < 4096; t += 64) smP[t] = 0.0f;
  __syncthreads();
  if (tid == 0) {
    const float r2 = 0.7071067811865476f;
    smP[0] = 1.0f;
    for (int k = 1; k <= 32; ++k) {
      if (k == 32) smP[32 * 64 + 32] = 1.0f;
      else { smP[k * 64 + k] = r2; smP[k * 64 + (64 - k)] = r2; }
    }
    for (int k = 1; k <= 31; ++k) {
      smP[(32 + k) * 64 + k] = r2;
      smP[(32 + k) * 64 + (64 - k)] = -r2;
    }
  }
  __syncthreads();
  for (int t = tid; t < 4096; t += 64) {          // T = P @ S
    int i = t >> 6, j = t & 63;
    float s = 0.0f;
    for (int k = 0; k < 64; ++k) s += smP[i * 64 + k] * smS[k * 64 + j];
    smT[t] = s;
  }
  __syncthreads();
  for (int t = tid; t < 4096; t += 64) {          // CS = T @ P^T (into smS)
    int i = t >> 6, j = t & 63;
    float s = 0.0f;
    for (int k = 0; k < 64; ++k) s += smT[i * 64 + k] * smP[j * 64 + k];
    smS[t] = s;
  }
  __syncthreads();
  for (int t = tid; t < NJ * NJ; t += 64) {       // diagonal block, V = I
    int i = t / NJ, j = t % NJ;
    A[t] = smS[(r0 + i) * 64 + (r0 + j)];
    V[t] = (i == j) ? 1.0f : 0.0f;
  }
  __syncthreads();
  for (int sweep = 0; sweep < 12; ++sweep) {      // cyclic Jacobi
    for (int p = 0; p < NJ - 1; ++p) {
      for (int q = p + 1; q < NJ; ++q) {
        if (tid == 0) {
          float apq = A[p * NJ + q];
          if (fabsf(apq) < 1e-12f) { rot_c = 1.0f; rot_s = 0.0f; }
          else {
            float th = (A[q * NJ + q] - A[p * NJ + p]) / (2.0f * apq);
            float tt = (th >= 0.0f ? 1.0f : -1.0f) / (fabsf(th) + sqrtf(th * th + 1.0f));
            float c = 1.0f / sqrtf(tt * tt + 1.0f);
            rot_c = c; rot_s = tt * c;
          }
        }
        __syncthreads();
        float c = rot_c, s = rot_s;
        if (s != 0.0f) {
          for (int i = tid; i < NJ; i += 64) {    // A <- A J, V <- V J
            float aip = A[i * NJ + p], aiq = A[i * NJ + q];
            A[i * NJ + p] = c * aip - s * aiq;
            A[i * NJ + q] = s * aip + c * aiq;
            float vip = V[i * NJ + p], viq = V[i * NJ + q];
            V[i * NJ + p] = c * vip - s * viq;
            V[i * NJ + q] = s * vip + c * viq;
          }
        }
        __syncthreads();
        if (s != 0.0f) {
          for (int j = tid; j < NJ; j += 64) {    // A <- J^T A
            float apj = A[p * NJ + j], aqj = A[q * NJ + j];
            A[p * NJ + j] = c * apj - s * aqj;
            A[q * NJ + j] = s * apj + c * aqj;
          }
        }
        __syncthreads();
      }
    }
  }
  if (tid == 0) {                                  // sort descending
    for (int j = 0; j < NJ; ++j) ord[j] = j;
    for (int a = 0; a < NJ; ++a) {
      int best = a;
      for (int b = a + 1; b < NJ; ++b)
        if (A[ord[b] * NJ + ord[b]] > A[ord[best] * NJ + ord[best]]) best = b;
      int tmp = ord[a]; ord[a] = ord[best]; ord[best] = tmp;
    }
  }
  __syncthreads();
  const int ncols = NJ;                            // E[:,col] = P-block^T @ V[:,ord[j]]
  for (int t = tid; t < 64 * ncols; t += 64) {
    int n = t / ncols, j = t % ncols;
    float s = 0.0f;
    for (int k = 0; k < NJ; ++k) s += smP[(r0 + k) * 64 + n] * V[k * NJ + ord[j]];
    int col = (blockIdx.x == 0) ? ((j < 32) ? 2 * j : 63) : (2 * j + 1);
    gE[n * 64 + col] = s;
  }
}

// ---------------------------------------------------------------------------
// Setup kernel 2: planar F_op = E diag(exp(-i pi/2 alpha_op idx)) E^T.
// ops: 0:+a1 1:+a2 2:+a3 3:-a3 4:-a2 5:-a1   idx = [0..62, 64]
// Layout per op: 4096 floats re-plane, then 4096 floats im-plane ([m][k]).
// ---------------------------------------------------------------------------
__global__ __launch_bounds__(256) void frft_build_F(
    const float* __restrict__ gE,
    const float* __restrict__ a1, const float* __restrict__ a2,
    const float* __restrict__ a3, float* __restrict__ gF) {
  int e = blockIdx.x * 256 + threadIdx.x;   // 0..4095
  int op = blockIdx.y;
  int m = e >> 6, n = e & 63;
  float av0 = *a1, av1 = *a2, av2 = *a3;
  float alpha;
  if (op == 0) alpha = av0; else if (op == 1) alpha = av1;
  else if (op == 2) alpha = av2; else if (op == 3) alpha = -av2;
  else if (op == 4) alpha = -av1; else alpha = -av0;
  float sre = 0.0f, sim = 0.0f;
  for (int k = 0; k < 64; ++k) {
    float idxk = (k == 63) ? 64.0f : (float)k;
    float b = 1.5707963267948966f * alpha * idxk;
    float w = gE[m * 64 + k] * gE[n * 64 + k];
    sre += w * cosf(b);
    sim -= w * sinf(b);
  }
  gF[op * 8192 + m * 64 + n] = sre;
  gF[op * 8192 + 4096 + m * 64 + n] = sim;
}

// ---------------------------------------------------------------------------
// Setup kernel 3: planar Wc and bias.
// ---------------------------------------------------------------------------
__global__ __launch_bounds__(256) void frft_pack_w(
    const float* __restrict__ wr, const float* __restrict__ wi,
    const float* __restrict__ br, const float* __restrict__ bi,
    float* __restrict__ gW, float* __restrict__ gB) {
  int t = blockIdx.x * 256 + threadIdx.x;
  if (t < 1024) { gW[t] = wr[t]; gW[1024 + t] = wi[t]; }
  if (t < 32) { gB[t] = br[t]; gB[32 + t] = bi[t]; }
}

// ---------------------------------------------------------------------------
// Main batched complex GEMM along a strided axis (planar re/im storage):
//   out[line, m] = sum_n in[line, n] * F[m, n]  (+ bias[m])
// base(line) = (line>>LSH)<<(LSH+log2 KD) + (line & (L-1)); element k at
// base + k*L, L = 1<<LSH (covers W/H/D/channel axes, all compile-time).
//
// Per wave: one 16-line x KD C-tile, K-steps of 4 via v_wmma_f32_16x16x4_f32.
// Complex MAC (f32 WMMA has no A/B NEG -> negate A_im fragment in VALU):
//   acc_re += A_re*B_re + (-A_im)*B_im ;  acc_im += A_re*B_im + A_im*B_re
// B fragments: single ds_load_b64 each from planar LDS F (no shuffles).
// A fragments: b32 loads land directly in the WMMA register pair.
// ---------------------------------------------------------------------------
template <int KD, int LSH, bool REAL_IN, bool BIAS, bool FINAL>
__global__ __launch_bounds__(256) void frft_gemm(
    const float* __restrict__ in_re, const float* __restrict__ in_im,
    float* __restrict__ out_a, float* __restrict__ out_b,
    const float* __restrict__ Fm, const float* __restrict__ bias) {
  constexpr int KSH = (KD == 64) ? 6 : 5;
  constexpr int MT = KD / 16;
  constexpr int L = 1 << LSH;
  __shared__ __attribute__((aligned(16))) float sF[2 * KD * KD];  // [re|im], [m][k]

#if __has_builtin(__builtin_amdgcn_tensor_load_to_lds)
  // Stage F via the Tensor Data Mover: one 1-D descriptor copies the whole
  // planar matrix (2*KD*KD dwords) global -> LDS. Issued once by wave 0.
  // clang-23 form: (v4u g0, v8i g1, v4i g2, v4i g3, v8i g4, i32 cpol).
  if (threadIdx.x < 32) {
    constexpr unsigned NE = 2u * KD * KD;                  // dwords
    unsigned lds = (unsigned)(unsigned long long)(void*)sF;
    unsigned long long ga = (unsigned long long)Fm;
    v4u g0;
    g0.x = 1u;                                             // count=1, no gather
    g0.y = lds;                                            // D#.lds_addr
    g0.z = (unsigned)ga;                                   // global_addr[31:0]
    g0.w = (unsigned)((ga >> 32) & 0x01FFFFFFull) | (2u << 30);  // [56:32] | type=2
    v8i g1;
    g1[0] = (int)(2u << 16);                               // data_size = 4B
    g1[1] = (int)((NE & 0xFFFFu) << 16);                   // tensor_dim0[15:0]
    g1[2] = (int)((NE >> 16) | (1u << 16));                // tensor_dim0[31:16] | tensor_dim1=1
    g1[3] = (int)(NE << 16);                               // tile_dim0 = NE
    g1[4] = 0;                                             // tile_dim1/2 unused
    g1[5] = (int)NE;                                       // tensor_dim0_stride
    g1[6] = 0;
    g1[7] = 0;
    v4i gz4; gz4.x = 0; gz4.y = 0; gz4.z = 0; gz4.w = 0;
    v8i gz8;
    gz8[0] = 0; gz8[1] = 0; gz8[2] = 0; gz8[3] = 0;
    gz8[4] = 0; gz8[5] = 0; gz8[6] = 0; gz8[7] = 0;
    __builtin_amdgcn_tensor_load_to_lds(g0, g1, gz4, gz4, gz8, 0);
    __builtin_amdgcn_s_wait_tensorcnt(0);
  }
  __syncthreads();
#else
  for (int t = threadIdx.x; t < 2 * KD * KD; t += 256) sF[t] = Fm[t];
  __syncthreads();
#endif

  const int lane = threadIdx.x & 31;
  const int wave = threadIdx.x >> 5;
  const int row = lane & 15;     // A: M index / B,C: N index
  const int hl = lane >> 4;      // lane-half: K offset {0,2} / C M-half
  const int tile = blockIdx.x * 8 + wave;

  const int line = tile * 16 + row;
  const int baseIn = ((line >> LSH) << (LSH + KSH)) + (line & (L - 1));

  const v8f vzero = {0, 0, 0, 0, 0, 0, 0, 0};
  v8f acc_re[MT], acc_im[MT];
#pragma unroll
  for (int mt = 0; mt < MT; ++mt) { acc_re[mt] = vzero; acc_im[mt] = vzero; }

  for (int ks = 0; ks < KD / 4; ++ks) {
    const int k0 = ks * 4 + hl * 2;
    v2f a_re, a_im, a_imn;
    a_re.x = in_re[baseIn + k0 * L];
    a_re.y = in_re[baseIn + (k0 + 1) * L];
    if constexpr (!REAL_IN) {
      a_im.x = in_im[baseIn + k0 * L];
      a_im.y = in_im[baseIn + (k0 + 1) * L];
      a_imn = -a_im;
    }

#pragma unroll
    for (int mt = 0; mt < MT; ++mt) {
      const int m = mt * 16 + row;
      v2f b_re = *(const v2f*)(sF + m * KD + k0);               // ds_load_b64
      v2f b_im = *(const v2f*)(sF + KD * KD + m * KD + k0);     // ds_load_b64

      acc_re[mt] = __builtin_amdgcn_wmma_f32_16x16x4_f32(
          false, a_re, false, b_re, (short)0, acc_re[mt], false, false);
      acc_im[mt] = __builtin_amdgcn_wmma_f32_16x16x4_f32(
          false, a_re, false, b_im, (short)0, acc_im[mt], false, false);
      if constexpr (!REAL_IN) {
        acc_re[mt] = __builtin_amdgcn_wmma_f32_16x16x4_f32(
            false, a_imn, false, b_im, (short)0, acc_re[mt], false, false);
        acc_im[mt] = __builtin_amdgcn_wmma_f32_16x16x4_f32(
            false, a_im, false, b_re, (short)0, acc_im[mt], false, false);
      }
    }
  }

  // C layout: VGPR r holds line (tile*16 + r + hl*8), column = row (+16*mt)
#pragma unroll
  for (int mt = 0; mt < MT; ++mt) {
    const int m = mt * 16 + row;
    float bre = 0.0f, bim = 0.0f;
    if constexpr (BIAS) { bre = bias[m]; bim = bias[KD + m]; }
#pragma unroll
    for (int r = 0; r < 8; ++r) {
      const int lo = tile * 16 + r + hl * 8;
      const int ob = ((lo >> LSH) << (LSH + KSH)) + (lo & (L - 1));
      if constexpr (FINAL) {
        ((float2*)out_a)[ob + m * L] =
            make_float2(acc_re[mt][r] + bre, acc_im[mt][r] + bim);
      } else {
        out_a[ob + m * L] = acc_re[mt][r] + bre;
        out_b[ob + m * L] = acc_im[mt][r] + bim;
      }
    }
  }
  (void)bias; (void)out_b; (void)in_im;
}

// ---------------------------------------------------------------------------
// Launcher. Inputs: x, alpha1, alpha2, alpha3, w_r, w_i, b_r, b_i.
// Output: complex64 tensor (2,32,64,64,64) as interleaved floats in d_out.
// ---------------------------------------------------------------------------
extern "C" void kernel_launch(void* const* d_in, const int* in_sizes, int n_in,
                              void* d_out, int out_size, void* d_ws, size_t ws_size,
                              hipStream_t stream) {
  (void)in_sizes; (void)n_in; (void)out_size; (void)ws_size;

  const float* x  = (const float*)d_in[0];
  const float* a1 = (const float*)d_in[1];
  const float* a2 = (const float*)d_in[2];
  const float* a3 = (const float*)d_in[3];
  const float* wr = (const float*)d_in[4];
  const float* wi = (const float*)d_in[5];
  const float* br = (const float*)d_in[6];
  const float* bi = (const float*)d_in[7];

  char* ws = (char*)d_ws;
  float* gF  = (float*)(ws + 0);          // 6 ops * 8192 floats, planar
  float* gW  = (float*)(ws + 196608);     // 2048 floats, planar
  float* gB  = (float*)(ws + 204800);     // 64 floats, planar
  float* gE  = (float*)(ws + 205056);     // 64*64 fp32
  float* buf = (float*)(ws + (1u << 20)); // planar ping buffer (re | im)
  float* oA  = (float*)d_out;             // planar pong buffer / final output
  const int P = 16777216;                 // plane size (elements)

  frft_build_basis<<<2, 64, 0, stream>>>(gE);
  frft_build_F<<<dim3(16, 6), 256, 0, stream>>>(gE, a1, a2, a3, gF);
  frft_pack_w<<<4, 256, 0, stream>>>(wr, wi, br, bi, gW, gB);

  // K=64 stages: 262144 lines -> 2048 blocks (8 waves x 16 lines), exact.
  // K=32 channel stage: 524288 lines -> 4096 blocks, exact. No divergence.
  frft_gemm<64, 0,  true,  false, false><<<2048, 256, 0, stream>>>(x,   nullptr, oA,  oA + P,  gF + 0 * 8192, nullptr); // W,+a1
  frft_gemm<64, 6,  false, false, false><<<2048, 256, 0, stream>>>(oA,  oA + P,  buf, buf + P, gF + 1 * 8192, nullptr); // H,+a2
  frft_gemm<64, 12, false, false, false><<<2048, 256, 0, stream>>>(buf, buf + P, oA,  oA + P,  gF + 2 * 8192, nullptr); // D,+a3
  frft_gemm<32, 18, false, true,  false><<<4096, 256, 0, stream>>>(oA,  oA + P,  buf, buf + P, gW, gB);                 // Wc+b
  frft_gemm<64, 12, false, false, false><<<2048, 256, 0, stream>>>(buf, buf + P, oA,  oA + P,  gF + 3 * 8192, nullptr); // D,-a3
  frft_gemm<64, 6,  false, false, false><<<2048, 256, 0, stream>>>(oA,  oA + P,  buf, buf + P, gF + 4 * 8192, nullptr); // H,-a2
  frft_gemm<64, 0,  false, false, true ><<<2048, 256, 0, stream>>>(buf, buf + P, oA,  nullptr, gF + 5 * 8192, nullptr); // W,-a1
}